// SAGE_5033701671208
// MI455X (gfx1250) — compile-verified
//
#include <hip/hip_runtime.h>
#include <math.h>

typedef __attribute__((ext_vector_type(2))) float v2f;
typedef __attribute__((ext_vector_type(8))) float v8f;

#define N0 1000000
#define N1 40960
#define N2 4096
#define E1 1024000
#define E2 40960
#define IN_C 100
#define HID 256
#define OUTC 47

// ---------------- utility ----------------
__global__ void zero_f32(float* __restrict__ p, long n) {
  long i = (long)blockIdx.x * blockDim.x + threadIdx.x;
  long stride = (long)gridDim.x * blockDim.x;
  for (; i < n; i += stride) p[i] = 0.0f;
}

// ---------------- layer-1 scatter: one block per edge, 100 features ----------
__global__ void scatter1(const float* __restrict__ x, const int* __restrict__ src,
                         const int* __restrict__ dst, float* __restrict__ agg,
                         float* __restrict__ cnt) {
  int e = blockIdx.x;
  int s = src[e];
  int d = dst[e];
  int t = threadIdx.x;
  if (t < IN_C) atomicAdd(&agg[(long)d * IN_C + t], x[(long)s * IN_C + t]);
  if (t == 0)   atomicAdd(&cnt[d], 1.0f);
}

// ---------------- layer-2 scatter: one block per edge, 256 features ----------
__global__ void scatter2(const float* __restrict__ h, const int* __restrict__ src,
                         const int* __restrict__ dst, float* __restrict__ agg,
                         float* __restrict__ cnt) {
  int e = blockIdx.x;
  int s = src[e];
  int d = dst[e];
  int t = threadIdx.x;
  atomicAdd(&agg[(long)d * HID + t], h[(long)s * HID + t]);
  if (t == 0) atomicAdd(&cnt[d], 1.0f);
}

// ---------------- layer 1 GEMM: h = relu((agg/cnt)@Wl + x@Wr + b) ------------
// One wave32 per 16x16 tile. f32 WMMA, K-step 4, K = 100 (25 steps).
__global__ void gemm1(const float* __restrict__ agg, const float* __restrict__ cnt,
                      const float* __restrict__ x,  const float* __restrict__ Wl,
                      const float* __restrict__ Wr, const float* __restrict__ b,
                      float* __restrict__ hout) {
  int wave = threadIdx.x >> 5;
  int lane = threadIdx.x & 31;
  int tile = blockIdx.x * 4 + wave;        // 40960 tiles total, exact multiple
  int ct = tile & 15;                       // 16 col tiles (HID/16)
  int rt = tile >> 4;                       // 2560 row tiles (N1/16)
  int row0 = rt * 16, col0 = ct * 16;
  int half = lane >> 4;                     // 0: K+0/1, 1: K+2/3 (A); rows of B likewise
  int ml = lane & 15;

  int arow = row0 + ml;                     // A-matrix row handled by this lane
  float rc = 1.0f / fmaxf(cnt[arow], 1.0f); // mean divisor (0-safe)
  const float* ap = agg + (long)arow * IN_C + 2 * half;
  const float* xp = x   + (long)arow * IN_C + 2 * half;
  int bn = col0 + ml;                       // B/D column handled by this lane

  v8f c = {};
  #pragma unroll 5
  for (int kb = 0; kb < IN_C; kb += 4) {
    v2f aa, ax, bl, br;
    aa.x = ap[kb]     * rc;  aa.y = ap[kb + 1] * rc;   // A[m][2*half+j] (ISA layout)
    ax.x = xp[kb];           ax.y = xp[kb + 1];
    int k0 = kb + 2 * half;                            // B row = 2*half + j
    bl.x = Wl[(long)k0 * HID + bn];  bl.y = Wl[(long)(k0 + 1) * HID + bn];
    br.x = Wr[(long)k0 * HID + bn];  br.y = Wr[(long)(k0 + 1) * HID + bn];
    c = __builtin_amdgcn_wmma_f32_16x16x4_f32(false, aa, false, bl, (short)0, c, false, false);
    c = __builtin_amdgcn_wmma_f32_16x16x4_f32(false, ax, false, br, (short)0, c, false, false);
  }
  float bias = b[bn];
  #pragma unroll
  for (int v = 0; v < 8; ++v) {
    int m = row0 + v + 8 * half;            // D layout: VGPR v -> M = v + 8*half
    float r = c[v] + bias;
    hout[(long)m * HID + bn] = r > 0.0f ? r : 0.0f;
  }
}

// ---------------- layer 2 GEMM: logits = (agg/cnt)@Wl + h@Wr + b -------------
// K = 256 (64 steps), N = 47 padded to 48 (3 col tiles), masked loads/stores.
__global__ void gemm2(const float* __restrict__ agg, const float* __restrict__ cnt,
                      const float* __restrict__ h,  const float* __restrict__ Wl,
                      const float* __restrict__ Wr, const float* __restrict__ b,
                      float* __restrict__ logits) {
  int wave = threadIdx.x >> 5;
  int lane = threadIdx.x & 31;
  int tile = blockIdx.x * 4 + wave;        // 768 tiles total, exact multiple
  int ct = tile % 3;                        // 3 col tiles (48/16)
  int rt = tile / 3;                        // 256 row tiles (N2/16)
  int row0 = rt * 16, col0 = ct * 16;
  int half = lane >> 4;
  int ml = lane & 15;

  int arow = row0 + ml;
  float rc = 1.0f / fmaxf(cnt[arow], 1.0f);
  const float* ap = agg + (long)arow * HID + 2 * half;
  const float* hp = h   + (long)arow * HID + 2 * half;
  int bn = col0 + ml;
  bool nok = bn < OUTC;

  v8f c = {};
  #pragma unroll 8
  for (int kb = 0; kb < HID; kb += 4) {
    v2f aa, ax, bl, br;
    aa.x = ap[kb]     * rc;  aa.y = ap[kb + 1] * rc;
    ax.x = hp[kb];           ax.y = hp[kb + 1];
    int k0 = kb + 2 * half;
    bl.x = nok ? Wl[(long)k0 * OUTC + bn] : 0.0f;
    bl.y = nok ? Wl[(long)(k0 + 1) * OUTC + bn] : 0.0f;
    br.x = nok ? Wr[(long)k0 * OUTC + bn] : 0.0f;
    br.y = nok ? Wr[(long)(k0 + 1) * OUTC + bn] : 0.0f;
    c = __builtin_amdgcn_wmma_f32_16x16x4_f32(false, aa, false, bl, (short)0, c, false, false);
    c = __builtin_amdgcn_wmma_f32_16x16x4_f32(false, ax, false, br, (short)0, c, false, false);
  }
  if (nok) {
    float bias = b[bn];
    #pragma unroll
    for (int v = 0; v < 8; ++v) {
      int m = row0 + v + 8 * half;
      logits[(long)m * OUTC + bn] = c[v] + bias;
    }
  }
}

// ---------------- row-wise log_softmax over 47 classes, 1 wave per row -------
__global__ void lsm_rows(const float* __restrict__ logits, float* __restrict__ out) {
  int lane = threadIdx.x & 31;
  int wave = threadIdx.x >> 5;
  int r = blockIdx.x * 8 + wave;            // blockDim 256 = 8 waves; 512 blocks
  const float* p = logits + (long)r * OUTC;
  float v0 = (lane < OUTC)      ? p[lane]      : -INFINITY;
  float v1 = (lane + 32 < OUTC) ? p[lane + 32] : -INFINITY;
  float m = fmaxf(v0, v1);
  #pragma unroll
  for (int i = 16; i > 0; i >>= 1) m = fmaxf(m, __shfl_xor(m, i, 32));
  float s = expf(v0 - m) + expf(v1 - m);    // exp(-inf)=0 handles the padding
  #pragma unroll
  for (int i = 16; i > 0; i >>= 1) s += __shfl_xor(s, i, 32);
  float lse = m + logf(s);
  float* q = out + (long)r * OUTC;
  if (lane < OUTC)      q[lane]      = v0 - lse;
  if (lane + 32 < OUTC) q[lane + 32] = v1 - lse;
}

extern "C" void kernel_launch(void* const* d_in, const int* in_sizes, int n_in,
                              void* d_out, int out_size, void* d_ws, size_t ws_size,
                              hipStream_t stream) {
  (void)in_sizes; (void)n_in; (void)out_size; (void)ws_size;
  // setup_inputs() order:
  const float* x    = (const float*)d_in[0];   // [N0, IN_C]
  const float* W1l  = (const float*)d_in[1];   // [IN_C, HID]
  const float* b1l  = (const float*)d_in[2];   // [HID]
  const float* W1r  = (const float*)d_in[3];   // [IN_C, HID]
  const float* W2l  = (const float*)d_in[4];   // [HID, OUTC]
  const float* b2l  = (const float*)d_in[5];   // [OUTC]
  const float* W2r  = (const float*)d_in[6];   // [HID, OUTC]
  const int*   src1 = (const int*)d_in[7];     // [E1]
  const int*   dst1 = (const int*)d_in[8];     // [E1]
  const int*   src2 = (const int*)d_in[9];     // [E2]
  const int*   dst2 = (const int*)d_in[10];    // [E2]
  float* out = (float*)d_out;                  // [N2, OUTC]

  // workspace layout (floats)
  float* w = (float*)d_ws;
  float* agg1   = w;                                   // N1*IN_C   = 4,096,000
  float* cnt1   = agg1 + (size_t)N1 * IN_C;            // N1        =    40,960
  float* agg2   = cnt1 + N1;                           // N2*HID    = 1,048,576
  float* cnt2   = agg2 + (size_t)N2 * HID;             // N2        =     4,096
  float* h      = cnt2 + N2;                           // N1*HID    = 10,485,760
  float* logits = h + (size_t)N1 * HID;                // N2*OUTC   =   192,512

  // 1) zero the aggregation region (agg1,cnt1,agg2,cnt2 are contiguous)
  long nz = (long)N1 * IN_C + N1 + (long)N2 * HID + N2;
  hipLaunchKernelGGL(zero_f32, dim3(2048), dim3(256), 0, stream, agg1, nz);

  // 2) layer-1 gather/scatter mean aggregation
  hipLaunchKernelGGL(scatter1, dim3(E1), dim3(128), 0, stream, x, src1, dst1, agg1, cnt1);

  // 3) layer-1 fused dual-GEMM + bias + ReLU  (40960 tiles / 4 waves per block)
  hipLaunchKernelGGL(gemm1, dim3((N1 / 16) * (HID / 16) / 4), dim3(128), 0, stream,
                     agg1, cnt1, x, W1l, W1r, b1l, h);

  // 4) layer-2 gather/scatter mean aggregation over h
  hipLaunchKernelGGL(scatter2, dim3(E2), dim3(HID), 0, stream, h, src2, dst2, agg2, cnt2);

  // 5) layer-2 fused dual-GEMM + bias  (768 tiles / 4 waves per block)
  hipLaunchKernelGGL(gemm2, dim3((N2 / 16) * 3 / 4), dim3(128), 0, stream,
                     agg2, cnt2, h, W2l, W2r, b2l, logits);

  // 6) log_softmax rows
  hipLaunchKernelGGL(lsm_rows, dim3(N2 / 8), dim3(256), 0, stream, logits, out);
}